// BiomechanicsLoss_kdtree_85916525789530
// MI455X (gfx1250) — compile-verified
//
#include <hip/hip_runtime.h>
#include <math.h>

// ---------------------------------------------------------------------------
// BiomechanicsLoss (brute-force NN + strain quadratic form), MI455X / gfx1250.
//
// NN search uses V_WMMA_F32_16X16X4_F32 (fp32 in / fp32 acc) to compute 16x16
// Gram-matrix tiles of w @ w^T — same precision as the fp32 reference.
// Problem is L2-resident (~600KB); the VALU min/argmin updates dominate, so
// the inner loop is stripped to: 2 loads + 1 WMMA + 8 fma + 8 cmp + 16 sel,
// with scalar (SALU) loop control:
//   * sq_i is loop-invariant per row -> added once after the scan
//   * self-diagonal tile (ct == rtile) is peeled out of the hot loop
//   * rtile forced wave-uniform via readfirstlane -> s_cmp/s_cbranch loop,
//     EXEC provably all-ones across the WMMA (ISA requirement)
// ---------------------------------------------------------------------------

typedef float v2f __attribute__((ext_vector_type(2)));
typedef float v8f __attribute__((ext_vector_type(8)));

#define BIGF 1e30f

// ---------------- prep: pad W to float4, precompute sq + masked key ---------
__global__ __launch_bounds__(256) void prep_kernel(
    const float* __restrict__ new_xyz, const float* __restrict__ gt_sdf,
    float* __restrict__ W4, float* __restrict__ sq, float* __restrict__ key,
    int n) {
  int i = blockIdx.x * blockDim.x + threadIdx.x;
  if (i >= n) return;
  float x = new_xyz[3 * i + 0];
  float y = new_xyz[3 * i + 1];
  float z = new_xyz[3 * i + 2];
  W4[4 * i + 0] = x;
  W4[4 * i + 1] = y;
  W4[4 * i + 2] = z;
  W4[4 * i + 3] = 0.0f;  // K padded to 4 for the 16x16x4 WMMA
  float s = x * x + y * y + z * z;
  sq[i] = s;
  // column mask folded into the distance term: d2 = sq_i + key_j - 2*dot
  key[i] = (gt_sdf[i] < 1e-8f) ? s : BIGF;
}

// One 16-column tile: WMMA dot tile + min/argmin update of (key_j - 2*dot).
// SELF=true additionally masks the j==i diagonal candidate (only the
// ct == rtile tile needs it; peeled so the hot loop carries no selects).
template <bool SELF>
__device__ __forceinline__ void scan_tile(int ct, int m, int half, int rs,
                                          const v2f* __restrict__ Wv,
                                          const float* __restrict__ key,
                                          v2f afrag, float best[8],
                                          int bidx[8]) {
  const int j = (ct << 4) + m;  // this lane's candidate column
  v2f bfrag = Wv[(unsigned)((j << 1) + half)];
  float kj = key[j];  // sq_j, or BIG if the column is masked

  v8f acc = {};
  // D = A * B + 0 : 16x16 tile of dot(w_i, w_j), fp32 exact
  acc = __builtin_amdgcn_wmma_f32_16x16x4_f32(
      /*neg_a=*/false, afrag, /*neg_b=*/false, bfrag,
      /*c_mod=*/(short)0, acc, /*reuse_a=*/false, /*reuse_b=*/false);

#pragma unroll
  for (int r = 0; r < 8; ++r) {
    float cand = __builtin_fmaf(acc[r], -2.0f, kj);  // key_j - 2*dot
    if (SELF && r == rs) cand = BIGF;                // exclude j == i
    if (cand < best[r]) { best[r] = cand; bidx[r] = j; }
  }
}

// ---------------- NN: one wave32 per 16-row tile ----------------------------
// A frag (16x4 f32, 2 VGPR): lane m=l&15 is row M=m; lanes<16 hold K=0,1,
// lanes>=16 hold K=2,3.  B frag (4x16 = W^T columns) has the identical
// per-lane float2 pattern indexed by the column base.  D (8 VGPR): VGPR r,
// lanes 0-15 -> (M=r, N=lane), lanes 16-31 -> (M=r+8, N=lane-16).
__global__ __launch_bounds__(256) void nn_kernel(
    const float* __restrict__ W4, const float* __restrict__ sq,
    const float* __restrict__ key, float* __restrict__ nnval,
    int* __restrict__ nnidx, int ntiles) {
  const int lane = threadIdx.x & 31;
  const int wave = threadIdx.x >> 5;
  // wave-uniform by construction; readfirstlane makes it an SGPR so the scan
  // loops compile to scalar s_cmp/s_cbranch control (EXEC untouched).
  const int rtile = __builtin_amdgcn_readfirstlane(
      blockIdx.x * (blockDim.x >> 5) + wave);
  if (rtile >= ntiles) return;

  const int m    = lane & 15;   // column of D held by this lane / row of A frag
  const int half = lane >> 4;   // K split (0,1 vs 2,3) and row-half of D
  const int rowBase = rtile << 4;

  const v2f* Wv = (const v2f*)W4;  // 2 v2f per padded point

  // A fragment: 2 consecutive f32 of padded row (rowBase+m), K offset 2*half
  v2f afrag = Wv[(unsigned)(((rowBase + m) << 1) + half)];

  // per-lane copies of sq for the 8 rows this lane owns in D (used at the end)
  float sqr[8];
#pragma unroll
  for (int r = 0; r < 8; ++r) sqr[r] = sq[rowBase + (half << 3) + r];

  // self-diagonal: lane's column m equals one of its rows iff rs in [0,8)
  const int rs = m - (half << 3);

  float best[8];
  int   bidx[8];
#pragma unroll
  for (int r = 0; r < 8; ++r) { best[r] = 3.0e38f; bidx[r] = 0; }

  // ascending j keeps argmin first-occurrence semantics; self tile peeled
  for (int ct = 0; ct < rtile; ++ct)
    scan_tile<false>(ct, m, half, rs, Wv, key, afrag, best, bidx);
  scan_tile<true>(rtile, m, half, rs, Wv, key, afrag, best, bidx);
  for (int ct = rtile + 1; ct < ntiles; ++ct)
    scan_tile<false>(ct, m, half, rs, Wv, key, afrag, best, bidx);

  // reduce (value,idx) lexicographically across the 16 lanes of each half;
  // preserves argmin first-occurrence semantics (min key <=> min d2).
#pragma unroll
  for (int r = 0; r < 8; ++r) {
    float v = best[r];
    int   id = bidx[r];
#pragma unroll
    for (int off = 1; off < 16; off <<= 1) {
      float vo = __shfl_xor(v, off, 16);
      int   io = __shfl_xor(id, off, 16);
      if (vo < v || (vo == v && io < id)) { v = vo; id = io; }
    }
    if (m == 0) {
      int row = rowBase + (half << 3) + r;
      nnval[row] = v + sqr[r];  // d2 = sq_i + (key_j - 2*dot)
      nnidx[row] = id;
    }
  }
}

// ---------------- per-row strain + quadratic form + block reduction ---------
struct CMat {
  float B[9];  // inverse of upper-left 3x3 block of Ci
  float d;     // inverse of shear diagonal 2(1+VP)/EP
};

__global__ __launch_bounds__(256) void loss_kernel(
    const float* __restrict__ new_xyz, const float* __restrict__ xyz,
    const float* __restrict__ gt_sdf, const float* __restrict__ nnval,
    const int* __restrict__ nnidx, float* __restrict__ partials, CMat C,
    int n) {
  __shared__ float sQ[256];
  __shared__ float sC[256];
  const int tid = threadIdx.x;
  const int i = blockIdx.x * blockDim.x + tid;

  float qsq = 0.0f, cnt = 0.0f;
  if (i < n) {
    float d2  = nnval[i];
    float nnd = sqrtf(fmaxf(d2, 0.0f));
    bool inside = gt_sdf[i] < 1e-8f;
    bool valid  = inside && (nnd > 1e-8f);
    cnt = valid ? 1.0f : 0.0f;
    int j = nnidx[i];

    float wix = new_xyz[3 * i], wiy = new_xyz[3 * i + 1], wiz = new_xyz[3 * i + 2];
    float wjx = new_xyz[3 * j], wjy = new_xyz[3 * j + 1], wjz = new_xyz[3 * j + 2];
    float mix = wix - xyz[3 * i], miy = wiy - xyz[3 * i + 1], miz = wiz - xyz[3 * i + 2];
    float mjx = wjx - xyz[3 * j], mjy = wjy - xyz[3 * j + 1], mjz = wjz - xyz[3 * j + 2];

    float du = valid ? (mjx - mix) : 0.0f;
    float dv = valid ? (mjy - miy) : 0.0f;
    float dw = valid ? (mjz - miz) : 0.0f;
    float dx = valid ? (wjx - wix + 1e-8f) : 1.0f;
    float dy = valid ? (wjy - wiy + 1e-8f) : 1.0f;
    float dz = valid ? (wjz - wiz + 1e-8f) : 1.0f;

    float e0 = du / dx;
    float e1 = dv / dy;
    float e2 = dw / dz;
    float e3 = (du / dy + dv / dx) * 0.5f;
    float e4 = (du / dz + dw / dx) * 0.5f;
    float e5 = (dw / dy + dv / dz) * 0.5f;

    // q = et . C . et   (C block-diagonal: 3x3 B, then d*I on shear terms)
    float t0 = C.B[0] * e0 + C.B[1] * e1 + C.B[2] * e2;
    float t1 = C.B[3] * e0 + C.B[4] * e1 + C.B[5] * e2;
    float t2 = C.B[6] * e0 + C.B[7] * e1 + C.B[8] * e2;
    float q = e0 * t0 + e1 * t1 + e2 * t2 + C.d * (e3 * e3 + e4 * e4 + e5 * e5);
    q = valid ? q : 0.0f;
    qsq = q * q;  // norm accumulates squares
  }

  sQ[tid] = qsq;
  sC[tid] = cnt;
  __syncthreads();
  for (int s = 128; s > 0; s >>= 1) {
    if (tid < s) {
      sQ[tid] += sQ[tid + s];
      sC[tid] += sC[tid + s];
    }
    __syncthreads();
  }
  if (tid == 0) {
    partials[2 * blockIdx.x + 0] = sQ[0];
    partials[2 * blockIdx.x + 1] = sC[0];
  }
}

// ---------------- deterministic final reduction -----------------------------
__global__ void finalize_kernel(const float* __restrict__ partials, int nblk,
                                float* __restrict__ out) {
  if (blockIdx.x == 0 && threadIdx.x == 0) {
    float qs = 0.0f, c = 0.0f;
    for (int k = 0; k < nblk; ++k) {
      qs += partials[2 * k + 0];
      c  += partials[2 * k + 1];
    }
    out[0] = sqrtf(qs) / c;
  }
}

// ---------------- host: 3x3 inverse in double (mirrors np.linalg.inv) ------
static void invert3x3(const double* A, double* inv) {
  double M[3][6];
  for (int r = 0; r < 3; ++r) {
    M[r][0] = A[3 * r + 0];
    M[r][1] = A[3 * r + 1];
    M[r][2] = A[3 * r + 2];
    M[r][3] = (r == 0);
    M[r][4] = (r == 1);
    M[r][5] = (r == 2);
  }
  for (int c = 0; c < 3; ++c) {
    int piv = c;
    for (int r = c + 1; r < 3; ++r)
      if (fabs(M[r][c]) > fabs(M[piv][c])) piv = r;
    if (piv != c)
      for (int k = 0; k < 6; ++k) {
        double t = M[c][k];
        M[c][k] = M[piv][k];
        M[piv][k] = t;
      }
    double p = M[c][c];
    for (int k = 0; k < 6; ++k) M[c][k] /= p;
    for (int r = 0; r < 3; ++r) {
      if (r == c) continue;
      double f = M[r][c];
      for (int k = 0; k < 6; ++k) M[r][k] -= f * M[c][k];
    }
  }
  for (int r = 0; r < 3; ++r)
    for (int c = 0; c < 3; ++c) inv[3 * r + c] = M[r][3 + c];
}

extern "C" void kernel_launch(void* const* d_in, const int* in_sizes, int n_in,
                              void* d_out, int out_size, void* d_ws,
                              size_t ws_size, hipStream_t stream) {
  const float* new_xyz = (const float*)d_in[0];  // (N,3) f32
  const float* xyz     = (const float*)d_in[1];  // (N,3) f32
  const float* gt_sdf  = (const float*)d_in[2];  // (N,)  f32
  const int n = in_sizes[2];                     // 12288 (multiple of 16)

  // workspace layout (floats from d_ws base, all 16B-aligned for n mult of 16)
  float* W4       = (float*)d_ws;        // n*4 floats
  float* sq       = W4 + 4 * (size_t)n;  // n floats
  float* key      = sq + n;              // n floats
  float* nnval    = key + n;             // n floats
  int*   nnidx    = (int*)(nnval + n);   // n ints
  float* partials = (float*)(nnidx + n); // 2*nblk floats

  prep_kernel<<<(n + 255) / 256, 256, 0, stream>>>(new_xyz, gt_sdf, W4, sq,
                                                   key, n);

  const int ntiles = n / 16;                       // 768 row tiles
  const int wavesPerBlock = 8;                     // 256 threads = 8 wave32
  const int nnBlocks = (ntiles + wavesPerBlock - 1) / wavesPerBlock;
  nn_kernel<<<nnBlocks, 256, 0, stream>>>(W4, sq, key, nnval, nnidx, ntiles);

  // C = inv(Ci): block-diagonal (3x3 block + 0.075*I on shear), double prec.
  const double EP = 0.21, VP = 0.4;
  const double A[9] = {1.0 / EP, -VP / EP, -VP / EP,
                       -VP / EP, 1.0 / EP, -VP / EP,
                       -VP,      -VP,      1.0 / EP};
  double Bi[9];
  invert3x3(A, Bi);
  CMat C;
  for (int k = 0; k < 9; ++k) C.B[k] = (float)Bi[k];
  C.d = (float)(EP / (2.0 * (1.0 + VP)));  // 0.075

  const int nblk = (n + 255) / 256;
  loss_kernel<<<nblk, 256, 0, stream>>>(new_xyz, xyz, gt_sdf, nnval, nnidx,
                                        partials, C, n);
  finalize_kernel<<<1, 32, 0, stream>>>(partials, nblk, (float*)d_out);
}